// MMDLossFunction_9603546874013
// MI455X (gfx1250) — compile-verified
//
#include <hip/hip_runtime.h>

typedef __attribute__((ext_vector_type(2))) float v2f;
typedef __attribute__((ext_vector_type(8))) float v8f;

#define NROWS 4096
#define DFEAT 256
#define MTILES 256            // 4096/16 tiles in M
#define NSTRIPS 64            // 4096/64 strips in N (each strip = 4 tiles of 16)
#define WAVES_PER_COMBO (MTILES * NSTRIPS)   // 16384
#define TOTAL_WAVES (3 * WAVES_PER_COMBO)    // 49152

// ---------------------------------------------------------------------------
// Pass 1: fp32 squared row norms of X and Y (8192 rows total)
// ---------------------------------------------------------------------------
__global__ __launch_bounds__(256) void row_norms_kernel(
    const float* __restrict__ X, const float* __restrict__ Y,
    float* __restrict__ a2X, float* __restrict__ a2Y) {
  int t = blockIdx.x * 256 + threadIdx.x;               // 0..8191
  const float* src = (t < NROWS) ? (X + (size_t)t * DFEAT)
                                 : (Y + (size_t)(t - NROWS) * DFEAT);
  float s = 0.0f;
#pragma unroll 4
  for (int i = 0; i < DFEAT; i += 4) {
    float4 v = *(const float4*)(src + i);
    s = fmaf(v.x, v.x, s);
    s = fmaf(v.y, v.y, s);
    s = fmaf(v.z, v.z, s);
    s = fmaf(v.w, v.w, s);
  }
  if (t < NROWS) a2X[t] = s; else a2Y[t - NROWS] = s;
}

// ---------------------------------------------------------------------------
// Pass 2: per-wave 16x64 Gram strip via V_WMMA_F32_16X16X4_F32, fused
// exp(-d2/2) epilogue and wave reduction. One partial per wave.
//
// f32 A 16x4 fragment layout (ISA 7.12.2):
//   lane l, elem j  ->  M = l%16, K = 2*(l/16) + j
// B 4x16 mirrors with N in place of M, so for a Gram matrix (B = rows^T of the
// same row-major storage) A and B fragments load identically.
// ---------------------------------------------------------------------------
__global__ __launch_bounds__(256) void mmd_tile_kernel(
    const float* __restrict__ X, const float* __restrict__ Y,
    const float* __restrict__ a2X, const float* __restrict__ a2Y,
    float* __restrict__ partials) {
  const int lane = threadIdx.x & 31;
  const int wid  = blockIdx.x * 8 + (threadIdx.x >> 5);

  const int combo  = wid >> 14;        // 0: XX, 1: XY, 2: YY
  const int rem    = wid & 16383;
  const int mtile  = rem >> 6;         // 0..255
  const int nstrip = rem & 63;         // 0..63

  const float* Am = (combo == 2) ? Y : X;
  const float* Bm = (combo == 0) ? X : Y;
  const float* a2 = (combo == 2) ? a2Y : a2X;
  const float* b2 = (combo == 0) ? a2X : a2Y;

  const int mlane = lane & 15;
  const int koff  = (lane >> 4) << 1;  // 0 or 2

  const float* ap = Am + (size_t)(mtile * 16  + mlane) * DFEAT + koff;
  const float* bp = Bm + (size_t)(nstrip * 64 + mlane) * DFEAT + koff;

  v8f acc0 = {}, acc1 = {}, acc2 = {}, acc3 = {};

#pragma unroll 8
  for (int kb = 0; kb < 64; ++kb) {
    v2f a   = *(const v2f*)(ap + kb * 4);
    v2f b0  = *(const v2f*)(bp + kb * 4);
    v2f b1  = *(const v2f*)(bp + 16 * DFEAT + kb * 4);
    v2f b2v = *(const v2f*)(bp + 32 * DFEAT + kb * 4);
    v2f b3  = *(const v2f*)(bp + 48 * DFEAT + kb * 4);
    acc0 = __builtin_amdgcn_wmma_f32_16x16x4_f32(false, a, false, b0,  (short)0, acc0, false, false);
    acc1 = __builtin_amdgcn_wmma_f32_16x16x4_f32(false, a, false, b1,  (short)0, acc1, false, false);
    acc2 = __builtin_amdgcn_wmma_f32_16x16x4_f32(false, a, false, b2v, (short)0, acc2, false, false);
    acc3 = __builtin_amdgcn_wmma_f32_16x16x4_f32(false, a, false, b3,  (short)0, acc3, false, false);
  }

  // C/D layout: VGPR r, lanes 0-15 -> M=r, N=lane; lanes 16-31 -> M=8+r, N=lane-16
  const int mbase = mtile * 16 + ((lane >> 4) << 3);
  const int ncol  = lane & 15;

  float arow[8];
#pragma unroll
  for (int r = 0; r < 8; ++r) arow[r] = a2[mbase + r];

  const v8f accs[4] = {acc0, acc1, acc2, acc3};
  float sum = 0.0f;
#pragma unroll
  for (int t = 0; t < 4; ++t) {
    float bn = b2[nstrip * 64 + t * 16 + ncol];
    v8f acc = accs[t];
#pragma unroll
    for (int r = 0; r < 8; ++r) {
      float d2 = arow[r] + bn - 2.0f * acc[r];
      d2 = fmaxf(d2, 0.0f);
      sum += __expf(-0.5f * d2);
    }
  }

  // wave32 reduction (deterministic; no float atomics)
#pragma unroll
  for (int off = 16; off > 0; off >>= 1)
    sum += __shfl_down(sum, off, 32);
  if (lane == 0) partials[wid] = sum;
}

// ---------------------------------------------------------------------------
// Pass 3: deterministic reduction of partials; combine the three combos.
// ---------------------------------------------------------------------------
__global__ __launch_bounds__(256) void finalize_kernel(
    const float* __restrict__ partials, float* __restrict__ out) {
  __shared__ float red0[256], red1[256], red2[256];
  int t = threadIdx.x;
  float s0 = 0.0f, s1 = 0.0f, s2 = 0.0f;
  for (int i = t; i < WAVES_PER_COMBO; i += 256) {
    s0 += partials[i];
    s1 += partials[WAVES_PER_COMBO + i];
    s2 += partials[2 * WAVES_PER_COMBO + i];
  }
  red0[t] = s0; red1[t] = s1; red2[t] = s2;
  __syncthreads();
  for (int off = 128; off > 0; off >>= 1) {
    if (t < off) {
      red0[t] += red0[t + off];
      red1[t] += red1[t + off];
      red2[t] += red2[t + off];
    }
    __syncthreads();
  }
  if (t == 0) {
    const float inv = 1.0f / (4096.0f * 4096.0f);
    out[0] = (red0[0] - 2.0f * red1[0] + red2[0]) * inv;
  }
}

extern "C" void kernel_launch(void* const* d_in, const int* in_sizes, int n_in,
                              void* d_out, int out_size, void* d_ws, size_t ws_size,
                              hipStream_t stream) {
  const float* X = (const float*)d_in[0];
  const float* Y = (const float*)d_in[1];

  float* ws       = (float*)d_ws;
  float* partials = ws;                         // 49152 floats
  float* a2X      = ws + TOTAL_WAVES;           // 4096 floats
  float* a2Y      = ws + TOTAL_WAVES + NROWS;   // 4096 floats

  row_norms_kernel<<<32, 256, 0, stream>>>(X, Y, a2X, a2Y);
  mmd_tile_kernel<<<TOTAL_WAVES / 8, 256, 0, stream>>>(X, Y, a2X, a2Y, partials);
  finalize_kernel<<<1, 256, 0, stream>>>(partials, (float*)d_out);
}